// InfoNCECauchySemiSupervised_8572754722865
// MI455X (gfx1250) — compile-verified
//
#include <hip/hip_runtime.h>
#include <hip/hip_bf16.h>

typedef __attribute__((ext_vector_type(2))) float v2f;
typedef __attribute__((ext_vector_type(8))) float v8f;

#define D   128      // feature dim
#define L   2048     // each sub-block (b_l = b_u = 2048)
#define AP  36       // LDS pitch for 32-wide K chunk (conflict-free: gcd(36,64)=4)

// Descriptor tables for the 10 Cauchy matrices.
// flags: bit0 = exclude diag from row/col sums (rowsum(Z - eye));
//        bits1-2 = mask mode (0 none, 1 m_ii, 2 m_ij);
//        bit3 = diag log (p1); bits4-5 = log weight into pl[] (1 or 2)
//  z:            u11   u22   u12  l11   l22   l12  u1l1 u1l2 u2l1 u2l2
__constant__ int c_xoff[10] = {2048, 6144, 2048,    0, 4096,    0, 2048, 2048, 6144, 6144};
__constant__ int c_yoff[10] = {2048, 6144, 6144,    0, 4096, 4096,    0, 4096,    0, 4096};
__constant__ int c_rowT[10] = {   0,    1,    0,    2,    3,    2,    0,    0,    1,    1};
__constant__ int c_colT[10] = {  -1,   -1,    1,   -1,   -1,    3,    2,    3,    2,    3};
__constant__ int c_flag[10] = {   1,    1,    8,   19,   19,   36,    0,    0,    0,    0};

__global__ __launch_bounds__(256) void k_zero(float* __restrict__ p, int n) {
    int i = blockIdx.x * 256 + threadIdx.x;
    if (i < n) p[i] = 0.0f;
}

// Row norms: one wave (32 lanes) per feature row; lane loads float4.
__global__ __launch_bounds__(256) void k_norms(const float* __restrict__ F,
                                               float* __restrict__ nrm) {
    int row  = blockIdx.x * 8 + (threadIdx.x >> 5);
    int lane = threadIdx.x & 31;
    float4 v = *(const float4*)&F[(size_t)row * D + lane * 4];
    float s = v.x * v.x + v.y * v.y + v.z * v.z + v.w * v.w;
    s += __shfl_xor(s, 1);  s += __shfl_xor(s, 2);
    s += __shfl_xor(s, 4);  s += __shfl_xor(s, 8);  s += __shfl_xor(s, 16);
    if (lane == 0) nrm[row] = s;
}

// Fused Cauchy-GEMM + reduction kernel. grid = (16,16,10), block = 256 (8 waves).
// Block tile 128x128, wave tile 32x64 (2x4 v8f accumulators), fp32 WMMA 16x16x4.
__global__ __launch_bounds__(256) void k_main(const float* __restrict__ F,
                                              const int* __restrict__ lab,
                                              const float* __restrict__ nrm,
                                              float* __restrict__ rs,
                                              float* __restrict__ pl,
                                              float* __restrict__ p1) {
    __shared__ float As[128 * AP];
    __shared__ float Bs[128 * AP];
    __shared__ float xn[128], yn[128];
    __shared__ int   labR[128], labC[128];

    const int z        = blockIdx.z;
    const int xOff     = c_xoff[z], yOff = c_yoff[z];
    const int rowT     = c_rowT[z], colT = c_colT[z];
    const int flag     = c_flag[z];
    const int exclDiag = flag & 1;
    const int maskMode = (flag >> 1) & 3;
    const int diagLog  = (flag >> 3) & 1;
    const int logW     = (flag >> 4) & 3;

    const int tileM = blockIdx.y * 128;
    const int tileN = blockIdx.x * 128;
    const int t    = threadIdx.x;
    const int lane = t & 31;
    const int w    = t >> 5;
    const int wm   = w & 3;        // 4 wave-rows of 32
    const int wn   = w >> 2;       // 2 wave-cols of 64
    const int fr   = lane & 15;    // fragment row (M for A, N for B)
    const int fg   = lane >> 4;    // half-wave group
    const int fk   = fg * 2;       // K offset within 4-step

    if (t < 128) {
        xn[t]   = nrm[xOff + tileM + t];
        yn[t]   = nrm[yOff + tileN + t];
        labR[t] = lab[tileM + t];   // masks always use labels[0:2048], matrix-local index
        labC[t] = lab[tileN + t];
    }

    v8f acc[2][4];
#pragma unroll
    for (int mi = 0; mi < 2; ++mi)
#pragma unroll
        for (int ni = 0; ni < 4; ++ni)
            acc[mi][ni] = (v8f){0.f, 0.f, 0.f, 0.f, 0.f, 0.f, 0.f, 0.f};

    for (int kc = 0; kc < D; kc += 32) {
        if (kc) __syncthreads();
        // Stage 128x32 A and B panels (coalesced float4 global loads).
#pragma unroll
        for (int it = 0; it < 4; ++it) {
            int i   = t + it * 256;            // 0..1023
            int row = i >> 3;
            int c4  = (i & 7) << 2;
            float4 va = *(const float4*)&F[(size_t)(xOff + tileM + row) * D + kc + c4];
            float4 vb = *(const float4*)&F[(size_t)(yOff + tileN + row) * D + kc + c4];
            *(float4*)&As[row * AP + c4] = va;
            *(float4*)&Bs[row * AP + c4] = vb;
        }
        __syncthreads();
#pragma unroll
        for (int kk = 0; kk < 32; kk += 4) {
            v2f a0 = *(const v2f*)&As[(wm * 32 +  0 + fr) * AP + kk + fk];
            v2f a1 = *(const v2f*)&As[(wm * 32 + 16 + fr) * AP + kk + fk];
            v2f b0 = *(const v2f*)&Bs[(wn * 64 +  0 + fr) * AP + kk + fk];
            v2f b1 = *(const v2f*)&Bs[(wn * 64 + 16 + fr) * AP + kk + fk];
            v2f b2 = *(const v2f*)&Bs[(wn * 64 + 32 + fr) * AP + kk + fk];
            v2f b3 = *(const v2f*)&Bs[(wn * 64 + 48 + fr) * AP + kk + fk];
            acc[0][0] = __builtin_amdgcn_wmma_f32_16x16x4_f32(false, a0, false, b0, (short)0, acc[0][0], false, false);
            acc[0][1] = __builtin_amdgcn_wmma_f32_16x16x4_f32(false, a0, false, b1, (short)0, acc[0][1], false, false);
            acc[0][2] = __builtin_amdgcn_wmma_f32_16x16x4_f32(false, a0, false, b2, (short)0, acc[0][2], false, false);
            acc[0][3] = __builtin_amdgcn_wmma_f32_16x16x4_f32(false, a0, false, b3, (short)0, acc[0][3], false, false);
            acc[1][0] = __builtin_amdgcn_wmma_f32_16x16x4_f32(false, a1, false, b0, (short)0, acc[1][0], false, false);
            acc[1][1] = __builtin_amdgcn_wmma_f32_16x16x4_f32(false, a1, false, b1, (short)0, acc[1][1], false, false);
            acc[1][2] = __builtin_amdgcn_wmma_f32_16x16x4_f32(false, a1, false, b2, (short)0, acc[1][2], false, false);
            acc[1][3] = __builtin_amdgcn_wmma_f32_16x16x4_f32(false, a1, false, b3, (short)0, acc[1][3], false, false);
        }
    }

    // ---- Epilogue: dot -> Cauchy, masked logs, row/col reductions ----
    // C/D layout: element (M = r + 8*fg + 16*mi + 32*wm, N = fr + 16*ni + 64*wn), VGPR r.
    float rowAcc[2][8], logAcc[2][8], colAcc[4];
#pragma unroll
    for (int mi = 0; mi < 2; ++mi)
#pragma unroll
        for (int r = 0; r < 8; ++r) { rowAcc[mi][r] = 0.f; logAcc[mi][r] = 0.f; }
#pragma unroll
    for (int ni = 0; ni < 4; ++ni) colAcc[ni] = 0.f;

#pragma unroll
    for (int mi = 0; mi < 2; ++mi) {
#pragma unroll
        for (int ni = 0; ni < 4; ++ni) {
#pragma unroll
            for (int r = 0; r < 8; ++r) {
                int ml = wm * 32 + mi * 16 + 8 * fg + r;
                int nl = wn * 64 + ni * 16 + fr;
                int gm = tileM + ml, gn = tileN + nl;
                float dot = acc[mi][ni][r];
                float d2  = fmaxf(xn[ml] + yn[nl] - 2.0f * dot, 0.0f);
                float c   = 1.0f / (1.0f + d2);
                float cc  = (exclDiag && gm == gn) ? (c - 1.0f) : c;  // rowsum(Z - eye)
                rowAcc[mi][r] += cc;
                colAcc[ni]    += cc;
                if (maskMode) {
                    bool eq = (labR[ml] == labC[nl]);
                    if (maskMode == 1) eq = eq && (gm != gn);         // m_ii
                    if (eq) logAcc[mi][r] += __logf(c);
                }
                if (diagLog && gm == gn) atomicAdd(p1, __logf(c));    // p1 diag term
            }
        }
    }

    // Row sums: butterfly over the 16 lanes sharing (r, fg); lanes fr==0 commit.
#pragma unroll
    for (int mi = 0; mi < 2; ++mi) {
#pragma unroll
        for (int r = 0; r < 8; ++r) {
            float v = rowAcc[mi][r];
            v += __shfl_xor(v, 1); v += __shfl_xor(v, 2);
            v += __shfl_xor(v, 4); v += __shfl_xor(v, 8);
            float wv = logAcc[mi][r];
            if (logW) {
                wv += __shfl_xor(wv, 1); wv += __shfl_xor(wv, 2);
                wv += __shfl_xor(wv, 4); wv += __shfl_xor(wv, 8);
            }
            if (fr == 0) {
                int gm = tileM + wm * 32 + mi * 16 + 8 * fg + r;
                if (rowT >= 0) atomicAdd(&rs[rowT * L + gm], v);
                if (logW)      atomicAdd(&pl[gm], (float)logW * wv);
            }
        }
    }
    // Col sums: per-lane sum over r (+mi) done above; fold the two half-wave row groups.
    if (colT >= 0) {
#pragma unroll
        for (int ni = 0; ni < 4; ++ni) {
            float v = colAcc[ni];
            v += __shfl_xor(v, 16);
            if (fg == 0) {
                int gn = tileN + wn * 64 + ni * 16 + fr;
                atomicAdd(&rs[colT * L + gn], v);
            }
        }
    }
}

// Final scalar assembly: denom2 from labels, log the n-sums, combine.
__global__ __launch_bounds__(1024) void k_final(const int* __restrict__ labels,
                                                const float* __restrict__ rs,
                                                const float* __restrict__ pl,
                                                const float* __restrict__ p1,
                                                float* __restrict__ out) {
    __shared__ int   labs[L];
    __shared__ float sp[32], sn[32];
    int t = threadIdx.x;
    for (int i = t; i < L; i += 1024) labs[i] = labels[i];
    __syncthreads();

    float accP = 0.f, accN = 0.f;
    for (int i = t; i < L; i += 1024) {
        int li = labs[i], cnt = 0;
        for (int j = 0; j < L; ++j) cnt += (labs[j] == li);
        float denom2 = (float)(2 * cnt - 1);          // sum(m_ii) + sum(m_ij) per row
        accP += pl[i] / denom2;                        // p2[i] + p3[i] numerators
        accN += __logf(rs[i])       + __logf(rs[L + i])
              + __logf(rs[2*L + i]) + __logf(rs[3*L + i]);   // n1..n4
    }
    for (int m = 1; m < 32; m <<= 1) {
        accP += __shfl_xor(accP, m);
        accN += __shfl_xor(accN, m);
    }
    int wv = t >> 5, lane = t & 31;
    if (lane == 0) { sp[wv] = accP; sn[wv] = accN; }
    __syncthreads();
    if (t == 0) {
        float P = 0.f, N = 0.f;
        for (int i = 0; i < 32; ++i) { P += sp[i]; N += sn[i]; }
        float pos = P / 8192.0f + p1[0] / 4096.0f;    // (Σp2+Σp3)/(2b) + p1
        float neg = N / 8192.0f;                      // mean of n1..n4
        out[0] = -(pos - neg);
    }
}

extern "C" void kernel_launch(void* const* d_in, const int* in_sizes, int n_in,
                              void* d_out, int out_size, void* d_ws, size_t ws_size,
                              hipStream_t stream) {
    (void)in_sizes; (void)n_in; (void)out_size; (void)ws_size;
    const float* F      = (const float*)d_in[0];   // features (8192 x 128) fp32
    const int*   labels = (const int*)d_in[1];     // labels (8192,) int32; only [0:2048] used
    float* out = (float*)d_out;

    float* ws  = (float*)d_ws;
    float* nrm = ws;            // 8192 row norms
    float* rs  = ws + 8192;     // 4 x 2048 accumulators for n1..n4
    float* pl  = ws + 16384;    // 2048 weighted masked-log sums (p2+p3 numerators)
    float* p1  = ws + 18432;    // 1 scalar (sum log diag z_u12)

    k_zero <<<(10241 + 255) / 256, 256, 0, stream>>>(rs, 10241);  // rs..p1 contiguous
    k_norms<<<1024, 256, 0, stream>>>(F, nrm);

    dim3 grid(16, 16, 10);
    k_main<<<grid, 256, 0, stream>>>(F, labels, nrm, rs, pl, p1);

    k_final<<<1, 1024, 0, stream>>>(labels, rs, pl, p1, out);
}